// GNN_atom_bond_34110630265243
// MI455X (gfx1250) — compile-verified
//
#include <hip/hip_runtime.h>
#include <hip/hip_bf16.h>

typedef float v2f __attribute__((ext_vector_type(2)));
typedef float v8f __attribute__((ext_vector_type(8)));
typedef int   v4i __attribute__((vector_size(16)));

#define B_     64
#define N_     128
#define K_     12
#define D_ATOM 256
#define D_NBR  320
#define D_NEW  256
#define H_     8
#define LN_EPS 1e-5f

#define APB        4                 // atoms per block
#define ROWS_E     (APB * K_)        // 48 rows in the e-GEMM (3 WMMA M-tiles)
#define SAE_STRIDE 324               // 320+4: bank-conflict-free, 16B-aligned rows
#define SAH_STRIDE 260               // 256+4: bank-conflict-free, 16B-aligned rows
#define SE_STRIDE  260               // 256+4 pad
#define SH_STRIDE  260

#if __has_builtin(__builtin_amdgcn_global_load_async_to_lds_b128) && \
    __has_builtin(__builtin_amdgcn_s_wait_asynccnt)
#define USE_ASYNC_LDS 1
#else
#define USE_ASYNC_LDS 0
#endif

#define AS1 __attribute__((address_space(1)))
#define AS3 __attribute__((address_space(3)))

__device__ __forceinline__ v8f wmma4(v2f a, v2f b, v8f c) {
  // V_WMMA_F32_16X16X4_F32 : D(16x16,f32) = A(16x4,f32) x B(4x16,f32) + C
  return __builtin_amdgcn_wmma_f32_16x16x4_f32(false, a, false, b, (short)0, c,
                                               false, false);
}

__global__ __launch_bounds__(128, 1)
void gatv2_fused_kernel(const float* __restrict__ atomf,   // (B,N,256)
                        const float* __restrict__ nbrf,    // (B,N,12,320)
                        const float* __restrict__ smask,   // (B,N,12,1) additive
                        const float* __restrict__ amask,   // (B,N,12,1) mult
                        const float* __restrict__ W_atom,  // (256,256)
                        const float* __restrict__ b_atom,  // (256)
                        const float* __restrict__ W_nbr,   // (320,256)
                        const float* __restrict__ b_nbr,   // (256)
                        const float* __restrict__ w_align, // (32,1)
                        const float* __restrict__ b_align, // (1)
                        const float* __restrict__ gamma,   // (256)
                        const float* __restrict__ beta,    // (256)
                        float* __restrict__ out)           // (B,N,256)
{
  __shared__ float sAe[ROWS_E * SAE_STRIDE];   // neighbor A operand, 48x320
  __shared__ float sAh[16 * SAH_STRIDE];       // atom A operand, 16x256 (rows 4..15 zero)
  __shared__ float sE[ROWS_E * SE_STRIDE];     // full e tile, 48x256
  __shared__ float sH[APB * SH_STRIDE];        // full h tile, 4x256
  __shared__ float sScore[ROWS_E * H_];        // logits -> attn, 48x8
  __shared__ float sCtx[APB * D_NEW];          // context, 4x256

  const int tid  = threadIdx.x;
  const int wave = tid >> 5;
  const int lane = tid & 31;
  const int lm   = lane & 15;   // M (A) / N (B,D) index within the 16-tile
  const int lh   = lane >> 4;   // K-half selector of the 16x16x4 op

  const int atom0 = blockIdx.x * APB;   // global atom index base

  // -------- kick off the big neighbor-tile copy (async HBM -> LDS) --------
  {
    const float* src = nbrf + (size_t)atom0 * K_ * D_NBR;  // 48 contiguous rows
#if USE_ASYNC_LDS
    AS3 float* dst = (AS3 float*)&sAe[0];
    for (int idx = tid; idx < ROWS_E * (D_NBR / 4); idx += 128) {
      const int r  = idx / (D_NBR / 4);
      const int c4 = (idx % (D_NBR / 4)) * 4;
      __builtin_amdgcn_global_load_async_to_lds_b128(
          (AS1 v4i*)(src + r * D_NBR + c4),
          (AS3 v4i*)(dst + r * SAE_STRIDE + c4), 0, 0);
    }
#else
    for (int idx = tid; idx < ROWS_E * (D_NBR / 4); idx += 128) {
      const int r  = idx / (D_NBR / 4);
      const int c4 = (idx % (D_NBR / 4)) * 4;
      const float4 v = *(const float4*)(src + r * D_NBR + c4);
      float* d = &sAe[r * SAE_STRIDE + c4];
      d[0] = v.x; d[1] = v.y; d[2] = v.z; d[3] = v.w;
    }
#endif
  }
  // -------- stage the small atom tile (sync) + zero padding rows --------
  {
    const float* src = atomf + (size_t)atom0 * D_ATOM;
    for (int idx = tid; idx < APB * (D_ATOM / 4); idx += 128) {
      const int r  = idx / (D_ATOM / 4);
      const int c4 = (idx % (D_ATOM / 4)) * 4;
      const float4 v = *(const float4*)(src + r * D_ATOM + c4);
      float* d = &sAh[r * SAH_STRIDE + c4];
      d[0] = v.x; d[1] = v.y; d[2] = v.z; d[3] = v.w;
    }
    for (int idx = tid; idx < (16 - APB) * D_ATOM; idx += 128) {
      const int r = APB + idx / D_ATOM;
      const int c = idx % D_ATOM;
      sAh[r * SAH_STRIDE + c] = 0.f;
    }
  }
  for (int idx = tid; idx < ROWS_E * H_; idx += 128) sScore[idx] = 0.f;
  __syncthreads();   // sAh/sScore visible; sAe still streaming (ASYNCcnt pending)

  // ==== Phase 0: h GEMM for all 4 slabs of this wave (overlaps async copy) ====
  {
    v8f acch[4];
    acch[0] = 0.f; acch[1] = 0.f; acch[2] = 0.f; acch[3] = 0.f;
#pragma unroll 2
    for (int kk = 0; kk < (D_ATOM / 4); ++kk) {
      const int col = kk * 4 + 2 * lh;
      const v2f ah = *(const v2f*)&sAh[lm * SAH_STRIDE + col];
#pragma unroll
      for (int i = 0; i < 4; ++i) {
        const int nc0 = (i * 4 + wave) * 16;
        v2f bh;
        bh.x = W_atom[(col    ) * D_NEW + nc0 + lm];
        bh.y = W_atom[(col + 1) * D_NEW + nc0 + lm];
        acch[i] = wmma4(ah, bh, acch[i]);
      }
    }
#pragma unroll
    for (int i = 0; i < 4; ++i) {
      const int nc0 = (i * 4 + wave) * 16;
      if (lh == 0) {
#pragma unroll
        for (int r = 0; r < APB; ++r)
          sH[r * SH_STRIDE + nc0 + lm] = acch[i][r];
      }
    }
  }
#if USE_ASYNC_LDS
  __builtin_amdgcn_s_wait_asynccnt(0);   // neighbor tile landed in LDS
#endif
  __syncthreads();

  // ============ Phase 1: e GEMM slabs + attention logits ============
  for (int i = 0; i < 4; ++i) {
    const int nt  = i * 4 + wave;      // 16-column slab owned by this wave
    const int nc0 = nt * 16;
    v8f acc0 = 0.f, acc1 = 0.f, acc2 = 0.f;
#pragma unroll 4
    for (int kk = 0; kk < (D_NBR / 4); ++kk) {
      const int col = kk * 4 + 2 * lh;
      const v2f a0 = *(const v2f*)&sAe[( 0 + lm) * SAE_STRIDE + col];
      const v2f a1 = *(const v2f*)&sAe[(16 + lm) * SAE_STRIDE + col];
      const v2f a2 = *(const v2f*)&sAe[(32 + lm) * SAE_STRIDE + col];
      v2f b;
      b.x = W_nbr[(col    ) * D_NEW + nc0 + lm];
      b.y = W_nbr[(col + 1) * D_NEW + nc0 + lm];
      acc0 = wmma4(a0, b, acc0);
      acc1 = wmma4(a1, b, acc1);
      acc2 = wmma4(a2, b, acc2);
    }
    // spill D tiles (VGPR r holds rows r and r+8) into the persistent e tile
#pragma unroll
    for (int r = 0; r < 8; ++r) {
      const int row = r + 8 * lh;
      sE[( 0 + row) * SE_STRIDE + nc0 + lm] = acc0[r];
      sE[(16 + row) * SE_STRIDE + nc0 + lm] = acc1[r];
      sE[(32 + row) * SE_STRIDE + nc0 + lm] = acc2[r];
    }
    __syncthreads();
    // partial logits for this slab: score[a,k,hd] += lrelu(h+e) . w_align
    const int hd  = nt >> 1;
    const int sub = nt & 1;
#pragma unroll
    for (int rr = 0; rr < 2; ++rr) {
      const int r = lane + rr * 32;
      if (r < ROWS_E) {
        const int a = r / K_;
        float accs = (sub == 0) ? b_align[0] : 0.f;
#pragma unroll
        for (int c = 0; c < 16; ++c) {
          float v = sH[a * SH_STRIDE + nc0 + c] + b_atom[nc0 + c]
                  + sE[r * SE_STRIDE + nc0 + c] + b_nbr[nc0 + c];
          v = (v > 0.f) ? v : 0.01f * v;                 // leaky_relu
          accs += v * w_align[sub * 16 + c];
        }
        atomicAdd(&sScore[r * H_ + hd], accs);           // ds_add_f32
      }
    }
    __syncthreads();
  }

  // ============ Phase 2: masked softmax over K per (atom, head) ============
  if (tid < APB * H_) {
    const int a  = tid >> 3;
    const int hd = tid & 7;
    const int g  = atom0 + a;
    float sc[K_];
    float mx = -3.4e38f;
#pragma unroll
    for (int k = 0; k < K_; ++k) {
      sc[k] = sScore[(a * K_ + k) * H_ + hd] + smask[g * K_ + k];
      mx = fmaxf(mx, sc[k]);
    }
    float sum = 0.f;
#pragma unroll
    for (int k = 0; k < K_; ++k) { sc[k] = __expf(sc[k] - mx); sum += sc[k]; }
    const float inv = 1.f / sum;
#pragma unroll
    for (int k = 0; k < K_; ++k)
      sScore[(a * K_ + k) * H_ + hd] = sc[k] * inv * amask[g * K_ + k];
  }
  __syncthreads();

  // ============ Phase 3: ctx = sum_k attn * (e + b_nbr), from cached e ============
  for (int idx = tid; idx < APB * D_NEW; idx += 128) {
    const int a   = idx >> 8;         // /256
    const int col = idx & 255;
    const int hd  = col >> 5;         // /32
    const float bn = b_nbr[col];
    float s = 0.f;
#pragma unroll
    for (int k = 0; k < K_; ++k)
      s += sScore[(a * K_ + k) * H_ + hd] *
           (sE[(a * K_ + k) * SE_STRIDE + col] + bn);
    sCtx[a * D_NEW + col] = s;
  }
  __syncthreads();

  // ============ Phase 4: LayerNorm, one wave32 per atom ============
  {
    const int a = wave;
    const int g = atom0 + a;
    float v[8];
    float s = 0.f;
#pragma unroll
    for (int j = 0; j < 8; ++j) { v[j] = sCtx[a * D_NEW + lane + j * 32]; s += v[j]; }
#pragma unroll
    for (int off = 16; off > 0; off >>= 1) s += __shfl_xor(s, off, 32);
    const float mu = s * (1.f / D_NEW);
    float q = 0.f;
#pragma unroll
    for (int j = 0; j < 8; ++j) { v[j] -= mu; q += v[j] * v[j]; }
#pragma unroll
    for (int off = 16; off > 0; off >>= 1) q += __shfl_xor(q, off, 32);
    const float rstd = rsqrtf(q * (1.f / D_NEW) + LN_EPS);
#pragma unroll
    for (int j = 0; j < 8; ++j) {
      const int col = lane + j * 32;
      out[(size_t)g * D_NEW + col] = v[j] * rstd * gamma[col] + beta[col];
    }
  }
}

extern "C" void kernel_launch(void* const* d_in, const int* in_sizes, int n_in,
                              void* d_out, int out_size, void* d_ws, size_t ws_size,
                              hipStream_t stream) {
  (void)in_sizes; (void)n_in; (void)out_size; (void)d_ws; (void)ws_size;
  const float* atomf   = (const float*)d_in[0];
  const float* nbrf    = (const float*)d_in[1];
  const float* smask   = (const float*)d_in[2];
  const float* amask   = (const float*)d_in[3];
  const float* W_atom  = (const float*)d_in[4];
  const float* b_atom  = (const float*)d_in[5];
  const float* W_nbr   = (const float*)d_in[6];
  const float* b_nbr   = (const float*)d_in[7];
  const float* w_align = (const float*)d_in[8];
  const float* b_align = (const float*)d_in[9];
  const float* gamma   = (const float*)d_in[10];
  const float* beta    = (const float*)d_in[11];
  float* out = (float*)d_out;

  const int blocks = (B_ * N_) / APB;   // 2048
  gatv2_fused_kernel<<<blocks, 128, 0, stream>>>(
      atomf, nbrf, smask, amask, W_atom, b_atom, W_nbr, b_nbr,
      w_align, b_align, gamma, beta, out);
}